// SignedGCN_36472862277897
// MI455X (gfx1250) — compile-verified
//
#include <hip/hip_runtime.h>

#define N_USERS 50000
#define N_FOODS 50000
#define NN (N_USERS + N_FOODS)     // 100000 nodes (divisible by 16)
#define H 128
#define F 64
#define E_EDGES 1000000

typedef __attribute__((ext_vector_type(2))) float v2f;
typedef __attribute__((ext_vector_type(8))) float v8f;

// ---------------- degree counting + inverse-clip ----------------
__global__ __launch_bounds__(256) void count_edges(const int* __restrict__ dst,
                                                   float* __restrict__ cnt, int n_edges) {
    int e = blockIdx.x * blockDim.x + threadIdx.x;
    if (e < n_edges) atomicAdd(&cnt[dst[e]], 1.0f);
}

__global__ __launch_bounds__(256) void invert_counts(float* __restrict__ c0,
                                                     float* __restrict__ c1, int n) {
    int i = blockIdx.x * blockDim.x + threadIdx.x;
    if (i < n) {
        c0[i] = 1.0f / fmaxf(c0[i], 1.0f);
        c1[i] = 1.0f / fmaxf(c1[i], 1.0f);
    }
}

// ---------------- edge scatter: S[dst] += X[src] (128 floats/row) ----------------
// One wave32 per edge; lane handles 4 consecutive floats (float4 gather + 4 f32 atomics;
// CDNA5 has no packed fp32 atomic, so this is the minimal exact-precision path).
__global__ __launch_bounds__(256) void scatter_rows(const int* __restrict__ src,
                                                    const int* __restrict__ dst,
                                                    const float* __restrict__ xa,
                                                    const float* __restrict__ xb, int split,
                                                    float* __restrict__ S, int n_edges) {
    int lane = threadIdx.x & 31;
    int e = (blockIdx.x * blockDim.x + threadIdx.x) >> 5;
    if (e >= n_edges) return;
    int s = src[e];
    int d = dst[e];
    const float* row = (s < split) ? (xa + (size_t)s * H)
                                   : (xb + (size_t)(s - split) * H);
    float4 v = *(const float4*)(row + lane * 4);
    float* out = S + (size_t)d * H + lane * 4;
    atomicAdd(out + 0, v.x);
    atomicAdd(out + 1, v.y);
    atomicAdd(out + 2, v.z);
    atomicAdd(out + 3, v.w);
}

// ---------------- layer 1 (one side): cols [col_base, col_base+64) of
// z = relu(agg@Wl + x@Wr + b). One wave per 16-row tile, 4 col tiles.
// Weights staged in LDS once per workgroup (all 8 waves share them).
// WMMA f32 16x16x4 fragment layouts (ISA 7.12.2):
//   A: lane -> row (lane&15); reg0=K=k+2*half, reg1=k+2*half+1
//   B: lane -> col (lane&15); reg0=row k+2*half, reg1=row k+2*half+1
//   C/D: reg v -> row v+8*half, col lane&15
__global__ __launch_bounds__(256) void layer1_wmma(
    const float* __restrict__ users, const float* __restrict__ items,
    const float* __restrict__ S, const float* __restrict__ inv,
    const float* __restrict__ Wl, const float* __restrict__ Wr,
    const float* __restrict__ bias, int col_base,
    float* __restrict__ z) {
    extern __shared__ float smem[];
    float* sWl = smem;            // H*F floats
    float* sWr = smem + H * F;    // H*F floats
    for (int i = threadIdx.x; i < H * F; i += 256) {
        sWl[i] = Wl[i];
        sWr[i] = Wr[i];
    }
    __syncthreads();               // all 8 waves reach this before any bounds-exit

    int lane = threadIdx.x & 31;
    int wave = (blockIdx.x * blockDim.x + threadIdx.x) >> 5;
    if (wave >= NN / 16) return;
    int base = wave * 16;
    int half = lane >> 4;
    int r = lane & 15;
    int my_row = base + r;

    const float* xrow = (my_row < N_USERS) ? (users + (size_t)my_row * H)
                                           : (items + (size_t)(my_row - N_USERS) * H);
    const float* srow = S + (size_t)my_row * H;
    float sc = inv[my_row];

    for (int tn = 0; tn < 4; ++tn) {
        int bc = tn * 16 + r;      // column within F

        v8f c = {};
        for (int k = 0; k < H; k += 4) {
            int ka = k + 2 * half;
            v2f a, b;
            // aggregated (mean-normalized) term
            a.x = srow[ka] * sc;
            a.y = srow[ka + 1] * sc;
            b.x = sWl[ka * F + bc];
            b.y = sWl[(ka + 1) * F + bc];
            c = __builtin_amdgcn_wmma_f32_16x16x4_f32(false, a, false, b, (short)0, c, false, false);
            // self term
            a.x = xrow[ka];
            a.y = xrow[ka + 1];
            b.x = sWr[ka * F + bc];
            b.y = sWr[(ka + 1) * F + bc];
            c = __builtin_amdgcn_wmma_f32_16x16x4_f32(false, a, false, b, (short)0, c, false, false);
        }
        float bv = bias[bc];
        float* orow = z + (size_t)base * H + col_base;
#pragma unroll
        for (int v = 0; v < 8; ++v) {
            int m = v + 8 * half;
            orow[(size_t)m * H + tn * 16 + r] = fmaxf(c[v] + bv, 0.0f);
        }
    }
}

// ---------------- layer 2 (one side) ----------------
// pos: out[:,0:64]  = relu([Sp*invp | Sn*invn](128) @ W2pl + z[:,0:64]  @ W2pr + b2p)
// neg: out[:,64:128]= relu([Sn*invn | Sp*invp](128) @ W2nl + z[:,64:128] @ W2nr + b2n)
// Slo/invlo cover K<64, Shi/invhi cover K>=64 (column index in S is the full-width k).
__global__ __launch_bounds__(256) void layer2_wmma(
    const float* __restrict__ z,
    const float* __restrict__ Slo, const float* __restrict__ Shi,
    const float* __restrict__ invlo, const float* __restrict__ invhi,
    const float* __restrict__ Wl, const float* __restrict__ Wr,
    const float* __restrict__ bias, int zoff, int col_base,
    float* __restrict__ out) {
    extern __shared__ float smem[];
    float* sWl = smem;            // (2F)*F floats
    float* sWr = smem + 2 * F * F; // F*F floats
    for (int i = threadIdx.x; i < 2 * F * F; i += 256) sWl[i] = Wl[i];
    for (int i = threadIdx.x; i < F * F; i += 256) sWr[i] = Wr[i];
    __syncthreads();

    int lane = threadIdx.x & 31;
    int wave = (blockIdx.x * blockDim.x + threadIdx.x) >> 5;
    if (wave >= NN / 16) return;
    int base = wave * 16;
    int half = lane >> 4;
    int r = lane & 15;
    int my_row = base + r;

    const float* zrow = z + (size_t)my_row * H + zoff;
    const float* slorow = Slo + (size_t)my_row * H;
    const float* shirow = Shi + (size_t)my_row * H;
    float sclo = invlo[my_row];
    float schi = invhi[my_row];

    for (int tn = 0; tn < 4; ++tn) {
        int bc = tn * 16 + r;

        v8f c = {};
        // K = 128 aggregated part (concat of two mean-normalized halves)
        for (int k = 0; k < H; k += 4) {
            int ka = k + 2 * half;
            const float* arow = (k < F) ? slorow : shirow;
            float sc = (k < F) ? sclo : schi;
            v2f a, b;
            a.x = arow[ka] * sc;
            a.y = arow[ka + 1] * sc;
            b.x = sWl[ka * F + bc];
            b.y = sWl[(ka + 1) * F + bc];
            c = __builtin_amdgcn_wmma_f32_16x16x4_f32(false, a, false, b, (short)0, c, false, false);
        }
        // K = 64 self part (z_p or z_n)
        for (int k = 0; k < F; k += 4) {
            int ka = k + 2 * half;
            v2f a, b;
            a.x = zrow[ka];
            a.y = zrow[ka + 1];
            b.x = sWr[ka * F + bc];
            b.y = sWr[(ka + 1) * F + bc];
            c = __builtin_amdgcn_wmma_f32_16x16x4_f32(false, a, false, b, (short)0, c, false, false);
        }
        float bv = bias[bc];
        float* orow = out + (size_t)base * H + col_base;
#pragma unroll
        for (int v = 0; v < 8; ++v) {
            int m = v + 8 * half;
            orow[(size_t)m * H + tn * 16 + r] = fmaxf(c[v] + bv, 0.0f);
        }
    }
}

extern "C" void kernel_launch(void* const* d_in, const int* in_sizes, int n_in,
                              void* d_out, int out_size, void* d_ws, size_t ws_size,
                              hipStream_t stream) {
    const float* users = (const float*)d_in[0];
    const float* items = (const float*)d_in[1];
    const float* w1pl = (const float*)d_in[2];
    const float* w1pr = (const float*)d_in[3];
    const float* b1p  = (const float*)d_in[4];
    const float* w1nl = (const float*)d_in[5];
    const float* w1nr = (const float*)d_in[6];
    const float* b1n  = (const float*)d_in[7];
    const float* w2pl = (const float*)d_in[8];
    const float* w2pr = (const float*)d_in[9];
    const float* b2p  = (const float*)d_in[10];
    const float* w2nl = (const float*)d_in[11];
    const float* w2nr = (const float*)d_in[12];
    const float* b2n  = (const float*)d_in[13];
    const int* pei = (const int*)d_in[14];   // (2, E): row0=src, row1=dst
    const int* nei = (const int*)d_in[15];
    const int* ps = pei;
    const int* pd = pei + E_EDGES;
    const int* ns = nei;
    const int* nd = nei + E_EDGES;

    // workspace layout (floats): invp[N] invn[N] Sp[N*H] Sn[N*H] z[N*H]
    float* invp = (float*)d_ws;
    float* invn = invp + NN;
    float* Sp   = invn + NN;
    float* Sn   = Sp + (size_t)NN * H;
    float* zbuf = Sn + (size_t)NN * H;
    float* out  = (float*)d_out;

    hipMemsetAsync(invp, 0, sizeof(float) * (size_t)2 * NN, stream);
    hipMemsetAsync(Sp, 0, sizeof(float) * (size_t)2 * NN * H, stream);  // Sp+Sn contiguous

    int eb = (E_EDGES + 255) / 256;
    count_edges<<<eb, 256, 0, stream>>>(pd, invp, E_EDGES);
    count_edges<<<eb, 256, 0, stream>>>(nd, invn, E_EDGES);
    invert_counts<<<(NN + 255) / 256, 256, 0, stream>>>(invp, invn, NN);

    int sb = (E_EDGES * 32 + 255) / 256;   // one wave per edge
    scatter_rows<<<sb, 256, 0, stream>>>(ps, pd, users, items, N_USERS, Sp, E_EDGES);
    scatter_rows<<<sb, 256, 0, stream>>>(ns, nd, users, items, N_USERS, Sn, E_EDGES);

    int waves = NN / 16;
    int wb = (waves * 32 + 255) / 256;
    size_t shm1 = (size_t)2 * H * F * sizeof(float);       // 64 KB
    layer1_wmma<<<wb, 256, shm1, stream>>>(users, items, Sp, invp,
                                           w1pl, w1pr, b1p, 0, zbuf);
    layer1_wmma<<<wb, 256, shm1, stream>>>(users, items, Sn, invn,
                                           w1nl, w1nr, b1n, F, zbuf);

    hipMemsetAsync(Sp, 0, sizeof(float) * (size_t)2 * NN * H, stream);
    scatter_rows<<<sb, 256, 0, stream>>>(ps, pd, zbuf, zbuf, NN, Sp, E_EDGES);
    scatter_rows<<<sb, 256, 0, stream>>>(ns, nd, zbuf, zbuf, NN, Sn, E_EDGES);

    size_t shm2 = (size_t)3 * F * F * sizeof(float);       // 48 KB
    layer2_wmma<<<wb, 256, shm2, stream>>>(zbuf, Sp, Sn, invp, invn,
                                           w2pl, w2pr, b2p, 0, 0, out);
    layer2_wmma<<<wb, 256, shm2, stream>>>(zbuf, Sn, Sp, invn, invp,
                                           w2nl, w2nr, b2n, F, F, out);
}